// MemRNN_55293408969197
// MI455X (gfx1250) — compile-verified
//
#include <hip/hip_runtime.h>
#include <math.h>

// Problem constants (reference: T,B,I,H = 512,64,512,512; I==H)
#define TT 512
#define BB 64
#define HH 512
#define KC 32   // K-chunk staged in LDS

typedef float v2f __attribute__((ext_vector_type(2)));
typedef float v8f __attribute__((ext_vector_type(8)));
typedef unsigned int v4u __attribute__((ext_vector_type(4)));
typedef int v4i __attribute__((ext_vector_type(4)));
typedef int v8i __attribute__((ext_vector_type(8)));

#if defined(__has_builtin)
#if __has_builtin(__builtin_amdgcn_tensor_load_to_lds)
#define HAVE_TDM 1
#endif
#endif

#ifdef HAVE_TDM
// ---------------------------------------------------------------------------
// TDM: DMA a 64-row x 32-col f32 tile (row stride 512 floats in memory) into
// LDS at byte offset `ldsOff`, with hardware padding of 1 DWORD after every
// 32 DWORDs (so LDS row stride = 33 floats, conflict-free for column access).
// D# layout per CDNA5 ISA ch.8 (group0: count/lds/global/type, group1: dims).
// ---------------------------------------------------------------------------
__device__ __forceinline__ void tdm_load_tile(unsigned ldsOff,
                                              const float* __restrict__ gptr) {
    unsigned long long ga = (unsigned long long)(uintptr_t)gptr;
    v4u g0;
    g0[0] = 1u;                                          // count=1, user D#
    g0[1] = ldsOff;                                      // lds_addr
    g0[2] = (unsigned)ga;                                // global_addr[31:0]
    g0[3] = (unsigned)((ga >> 32) & 0x1FFFFFFull)        // global_addr[56:32]
          | 0x80000000u;                                 // type=2 ("image")
    v8i g1;
    g1[0] = 0x01120000;        // data_size=4B | pad_enable | pad_interval=32dw (pad 1dw)
    g1[1] = (int)(32u << 16);  // tensor_dim0 = 32 (bits 79:48 -> low half here)
    g1[2] = (int)(64u << 16);  // tensor_dim1 = 64 (bits 95:80)
    g1[3] = (int)(32u << 16);  // tile_dim0 = 32 (bits 127:112)
    g1[4] = 64;                // tile_dim1 = 64 (bits 143:128)
    g1[5] = HH;                // tensor_dim0_stride = 512 (bits 191:160)
    g1[6] = 0;
    g1[7] = 0;
    v4i gz = {0, 0, 0, 0};
#if defined(__clang_major__) && (__clang_major__ >= 23)
    v8i gz8 = {0, 0, 0, 0, 0, 0, 0, 0};
    __builtin_amdgcn_tensor_load_to_lds(g0, g1, gz, gz, gz8, 0);
#else
    __builtin_amdgcn_tensor_load_to_lds(g0, g1, gz, gz, 0);
#endif
}
#endif // HAVE_TDM

// ---------------------------------------------------------------------------
// Generic small-M GEMM: out[64 x 512] = act( A1[64x512] @ W1[512x512]^T
//                                          (+ A2[64x512] @ W2[512x512]^T)
//                                          (+ bias[512]) )
// 256 threads = 8 waves; wave w computes a 16x32 C strip (2 accumulators
// sharing one A fragment):
//   M = (w&3)*16 , N = blockIdx.x*64 + (w>>2)*32 + {0,16}
// Double-buffered LDS chunks, TDM-DMA'd (TENSORcnt) when available.
// Uses V_WMMA_F32_16X16X4_F32.
// ---------------------------------------------------------------------------
__global__ __launch_bounds__(256) void gemm64(const float* __restrict__ A1,
                                              const float* __restrict__ W1,
                                              const float* __restrict__ A2,
                                              const float* __restrict__ W2,
                                              const float* __restrict__ bias,
                                              float* __restrict__ outPre,
                                              float* __restrict__ outTanh) {
    __shared__ float Asm[2][64][KC + 1];   // +1 pad matches TDM pad feature
    __shared__ float Wsm[2][64][KC + 1];

    const int tid  = threadIdx.x;
    const int lane = tid & 31;
    const int wave = tid >> 5;
    const int mT   = (wave & 3) * 16;       // wave's M tile origin
    const int nP   = (wave >> 2) * 32;      // wave's N strip origin (within 64)
    const int n0   = blockIdx.x * 64;       // block's N origin

    v8f acc0 = {}, acc1 = {};

    const int nc = (A2 == nullptr) ? (HH / KC) : (2 * HH / KC);  // 16 or 32 chunks

#ifdef HAVE_TDM
    // ---- TDM double-buffered pipeline ----
    if (tid == 0) {
        // prologue: chunk 0 -> buffer 0
        tdm_load_tile((unsigned)(uintptr_t)&Asm[0][0][0], A1);
        tdm_load_tile((unsigned)(uintptr_t)&Wsm[0][0][0], W1 + (size_t)n0 * HH);
        __builtin_amdgcn_s_wait_tensorcnt(0);
    }
    __syncthreads();

    for (int j = 0; j < nc; ++j) {
        const int p = j & 1;
        if (tid == 0 && j + 1 < nc) {       // prefetch next chunk into buf 1-p
            const int jn = j + 1;
            const float* A = (jn >= HH / KC) ? A2 : A1;
            const float* W = (jn >= HH / KC) ? W2 : W1;
            const int k0 = (jn & (HH / KC - 1)) * KC;
            tdm_load_tile((unsigned)(uintptr_t)&Asm[1 - p][0][0], A + k0);
            tdm_load_tile((unsigned)(uintptr_t)&Wsm[1 - p][0][0],
                          W + (size_t)n0 * HH + k0);
        }
        const int mrow  = mT + (lane & 15);
        const int nrow  = nP + (lane & 15);
        const int ksel  = (lane >> 4) << 1; // lanes 0-15 -> K 0,1 ; 16-31 -> K 2,3
#pragma unroll
        for (int kk = 0; kk < KC; kk += 4) {
            v2f a, b0, b1;
            a.x  = Asm[p][mrow][kk + ksel + 0];
            a.y  = Asm[p][mrow][kk + ksel + 1];
            b0.x = Wsm[p][nrow][kk + ksel + 0];
            b0.y = Wsm[p][nrow][kk + ksel + 1];
            b1.x = Wsm[p][nrow + 16][kk + ksel + 0];
            b1.y = Wsm[p][nrow + 16][kk + ksel + 1];
            acc0 = __builtin_amdgcn_wmma_f32_16x16x4_f32(
                false, a, false, b0, (short)0, acc0, false, false);
            acc1 = __builtin_amdgcn_wmma_f32_16x16x4_f32(
                false, a, false, b1, (short)0, acc1, false, false);
        }
        if (tid == 0 && j + 1 < nc)
            __builtin_amdgcn_s_wait_tensorcnt(0);   // next buffer landed
        __syncthreads();                            // publish / allow overwrite
    }
#else
    // ---- fallback: cooperative synchronous staging ----
    for (int j = 0; j < nc; ++j) {
        const float* A = (j >= HH / KC) ? A2 : A1;
        const float* W = (j >= HH / KC) ? W2 : W1;
        const int k0 = (j & (HH / KC - 1)) * KC;
        for (int i = tid; i < 64 * KC; i += 256) {
            int r = i >> 5, c = i & (KC - 1);
            Asm[0][r][c] = A[r * HH + k0 + c];
        }
        for (int i = tid; i < 64 * KC; i += 256) {
            int r = i >> 5, c = i & (KC - 1);
            Wsm[0][r][c] = W[(n0 + r) * HH + k0 + c];
        }
        __syncthreads();
        const int mrow = mT + (lane & 15);
        const int nrow = nP + (lane & 15);
        const int ksel = (lane >> 4) << 1;
#pragma unroll
        for (int kk = 0; kk < KC; kk += 4) {
            v2f a, b0, b1;
            a.x  = Asm[0][mrow][kk + ksel + 0];
            a.y  = Asm[0][mrow][kk + ksel + 1];
            b0.x = Wsm[0][nrow][kk + ksel + 0];
            b0.y = Wsm[0][nrow][kk + ksel + 1];
            b1.x = Wsm[0][nrow + 16][kk + ksel + 0];
            b1.y = Wsm[0][nrow + 16][kk + ksel + 1];
            acc0 = __builtin_amdgcn_wmma_f32_16x16x4_f32(
                false, a, false, b0, (short)0, acc0, false, false);
            acc1 = __builtin_amdgcn_wmma_f32_16x16x4_f32(
                false, a, false, b1, (short)0, acc1, false, false);
        }
        __syncthreads();
    }
#endif

    // Epilogue. C layout: VGPR r, lane l -> M = mT + r + (l>=16 ? 8 : 0),
    //                                      N = col0 / col1
    const int col0 = n0 + nP + (lane & 15);
    const int col1 = col0 + 16;
    const float bv0 = bias ? bias[col0] : 0.0f;
    const float bv1 = bias ? bias[col1] : 0.0f;
    const int rbase = mT + ((lane >> 4) << 3);
#pragma unroll
    for (int r = 0; r < 8; ++r) {
        const int row = rbase + r;
        const float pre0 = acc0[r] + bv0;
        const float pre1 = acc1[r] + bv1;
        if (outPre) {
            outPre[row * HH + col0] = pre0;
            outPre[row * HH + col1] = pre1;
        }
        if (outTanh) {
            outTanh[row * HH + col0] = tanhf(pre0);
            outTanh[row * HH + col1] = tanhf(pre1);
        }
    }
}

// ---------------------------------------------------------------------------
// Attention energies: e[b][tau] = sum_h tanh(vast[b,h] + uah[tau,b,h]) * v[h]
// One wave per (tau, b) pair; float4 (b128) lane-strided H reduction +
// shfl-xor wave reduce. e stored as [B][T] so the softmax reads contiguously.
// ---------------------------------------------------------------------------
__global__ __launch_bounds__(256) void energy_kernel(const float* __restrict__ vast,
                                                     const float* __restrict__ uah,
                                                     const float* __restrict__ v,
                                                     float* __restrict__ e,
                                                     int t) {
    const int lane = threadIdx.x & 31;
    const int pair = blockIdx.x * 8 + (threadIdx.x >> 5);
    if (pair >= t * BB) return;
    const int tau = pair >> 6;          // pair / 64
    const int b   = pair & 63;

    const float4* __restrict__ vp = (const float4*)(vast + (size_t)b * HH);
    const float4* __restrict__ up =
        (const float4*)(uah + ((size_t)tau * BB + b) * HH);
    const float4* __restrict__ vv = (const float4*)v;

    float s = 0.0f;
#pragma unroll
    for (int i = lane; i < HH / 4; i += 32) {
        float4 a = vp[i], u = up[i], w = vv[i];
        s += tanhf(a.x + u.x) * w.x;
        s += tanhf(a.y + u.y) * w.y;
        s += tanhf(a.z + u.z) * w.z;
        s += tanhf(a.w + u.w) * w.w;
    }

#pragma unroll
    for (int off = 16; off > 0; off >>= 1)
        s += __shfl_xor(s, off, 32);

    if (lane == 0) e[(size_t)b * TT + tau] = s;
}

// ---------------------------------------------------------------------------
// Per-b softmax over tau in [0,t), context ct = sum al*mem, and
// st_new = 0.5*(mem[t-1] + attn*ct).  One block per b; 256 threads, each
// thread owns one float2 column pair (b64 loads on the streamed memory).
// ---------------------------------------------------------------------------
__global__ __launch_bounds__(256) void attend_kernel(const float* __restrict__ e,
                                                     const float* __restrict__ mem,
                                                     const float* __restrict__ attnp,
                                                     float* __restrict__ st,
                                                     int t) {
    __shared__ float al[TT];
    __shared__ float red[256];
    const int b   = blockIdx.x;
    const int tid = threadIdx.x;

    // --- max ---
    float m = -3.0e38f;
    for (int tau = tid; tau < t; tau += 256) {
        float ev = e[(size_t)b * TT + tau];
        al[tau] = ev;
        m = fmaxf(m, ev);
    }
    red[tid] = m;
    __syncthreads();
    for (int s = 128; s > 0; s >>= 1) {
        if (tid < s) red[tid] = fmaxf(red[tid], red[tid + s]);
        __syncthreads();
    }
    m = red[0];
    __syncthreads();

    // --- exp + sum ---
    float ssum = 0.0f;
    for (int tau = tid; tau < t; tau += 256) {
        float ex = __expf(al[tau] - m);
        al[tau] = ex;
        ssum += ex;
    }
    red[tid] = ssum;
    __syncthreads();
    for (int s = 128; s > 0; s >>= 1) {
        if (tid < s) red[tid] += red[tid + s];
        __syncthreads();
    }
    const float inv = 1.0f / red[0];
    __syncthreads();

    // --- context + st_new (one float2 column pair per thread) ---
    const float a = attnp[0];
    const float2* __restrict__ mb2 =
        (const float2*)mem + (size_t)b * (HH / 2) + tid;
    const size_t tstride = (size_t)BB * HH / 2;   // float2 units per tau
    float accx = 0.0f, accy = 0.0f;
#pragma unroll 4
    for (int tau = 0; tau < t; ++tau) {
        const float w = al[tau];
        const float2 mv = mb2[(size_t)tau * tstride];
        accx = fmaf(w, mv.x, accx);
        accy = fmaf(w, mv.y, accy);
    }
    const float2 mprev = mb2[(size_t)(t - 1) * tstride];
    float2 r;
    r.x = 0.5f * (mprev.x + a * accx * inv);
    r.y = 0.5f * (mprev.y + a * accy * inv);
    ((float2*)st)[(size_t)b * (HH / 2) + tid] = r;
}

// ---------------------------------------------------------------------------
// Host-side orchestration (graph-capture safe: only kernel launches on stream)
// ---------------------------------------------------------------------------
extern "C" void kernel_launch(void* const* d_in, const int* in_sizes, int n_in,
                              void* d_out, int out_size, void* d_ws, size_t ws_size,
                              hipStream_t stream) {
    (void)in_sizes; (void)n_in; (void)out_size; (void)ws_size;
    const float* x    = (const float*)d_in[0];   // [T,B,I]
    const float* attn = (const float*)d_in[1];   // [1]
    const float* Uw   = (const float*)d_in[2];   // [H,I]
    const float* Ub   = (const float*)d_in[3];   // [H]
    const float* Vw   = (const float*)d_in[4];   // [H,H]
    const float* Uaw  = (const float*)d_in[5];   // [H,H]
    const float* Vaw  = (const float*)d_in[6];   // [H,H]
    const float* v    = (const float*)d_in[7];   // [H]

    float* out = (float*)d_out;                  // [T,B,H] == mem
    float* ws  = (float*)d_ws;
    float* uah  = ws;                                        // T*B*H
    float* st   = uah + (size_t)TT * BB * HH;                // B*H (carry)
    float* vast = st + (size_t)BB * HH;                      // B*H
    float* e    = vast + (size_t)BB * HH;                    // B*T

    dim3 gblk(8), blk(256);   // 8 blocks x 64 N-cols = 512

    // step 0: h0_pre = x[0]@Uw.T + Ub -> st ; h0 = tanh -> out[0] (=mem[0])
    gemm64<<<gblk, blk, 0, stream>>>(x, Uw, nullptr, nullptr, Ub, st, out);
    // uah[0] = h0 @ Uaw.T
    gemm64<<<gblk, blk, 0, stream>>>(out, Uaw, nullptr, nullptr, nullptr, uah, nullptr);

    for (int t = 1; t < TT; ++t) {
        // vast = st @ Vaw.T   (st is the carry from the previous step)
        gemm64<<<gblk, blk, 0, stream>>>(st, Vaw, nullptr, nullptr, nullptr, vast, nullptr);
        // energies over tau < t
        const int pairs = t * BB;
        energy_kernel<<<dim3((pairs + 7) / 8), blk, 0, stream>>>(vast, uah, v, e, t);
        // softmax + context + st_new (overwrites st)
        attend_kernel<<<dim3(BB), blk, 0, stream>>>(e, out, attn, st, t);
        // h = tanh(x[t]@Uw.T + st_new@Vw.T + Ub) -> out[t] (=mem[t])
        gemm64<<<gblk, blk, 0, stream>>>(x + (size_t)t * BB * HH, Uw,
                                         st, Vw, Ub,
                                         nullptr, out + (size_t)t * BB * HH);
        // uah[t] = h @ Uaw.T
        gemm64<<<gblk, blk, 0, stream>>>(out + (size_t)t * BB * HH, Uaw,
                                         nullptr, nullptr, nullptr,
                                         uah + (size_t)t * BB * HH, nullptr);
    }
}